// TorchInterp23_14568529068434
// MI455X (gfx1250) — compile-verified
//
#include <hip/hip_runtime.h>

// CDNA5 / gfx1250 fused 4x-upsampling separable polyphase filter.
// out[4m+p] = sum_t c_p[t] * x[m + t - 5] along W then H, with
// c_p[t] = BASE[4t+3-p]; each pass is a banded-Toeplitz matmul chain on
// V_WMMA_F32_16X16X4_F32. A-fragments are precomputed per (phase,lane) at
// compile time and fetched with b128 loads (no divergent coefficient loads).

typedef float v2f __attribute__((ext_vector_type(2)));
typedef float v8f __attribute__((ext_vector_type(8)));

#define HALO 28   // 16-sample tile + 11-tap halo (27) padded to 28 = 7*K4
#define XSTR 29   // LDS stride for input tile (coprime with 64 banks)
#define ISTR 65   // LDS stride for intermediate (coprime with 64 banks)

// Per-lane WMMA A-fragments of the banded Toeplitz F_p (16x28):
//   A[M][Kglob] = c_p[Kglob - M],  lane holds M = lane&15, K pair base = 2*(lane>>4)
// AFRAG.v[p][lane][2q+i] = value of A for K-chunk q, element i. 64B/lane row.
struct alignas(64) AF { float v[4][32][16]; };

constexpr AF make_af() {
  AF a{};
  const float BASE[45] = {
    -0.00463495665f, -0.00363442646f, 3.84904063e-18f, 0.00576678319f,
    0.0108358664f, 0.010198079f, -9.31747402e-18f, -0.0175033181f,
    -0.0317660068f, -0.0284531643f, 1.85181518e-17f, 0.0442450253f,
    0.0771733386f, 0.067055491f, -2.85299239e-17f, -0.101548683f,
    -0.178708388f, -0.160004642f, 3.61741232e-17f, 0.287940558f,
    0.625431459f, 0.8970676f, 1.00107877f, 0.8970676f, 0.625431459f,
    0.287940558f, 3.61741232e-17f, -0.160004642f, -0.178708388f,
    -0.101548683f, -2.85299239e-17f, 0.067055491f, 0.0771733386f,
    0.0442450253f, 1.85181518e-17f, -0.0284531643f, -0.0317660068f,
    -0.0175033181f, -9.31747402e-18f, 0.010198079f, 0.0108358664f,
    0.00576678319f, 3.84904063e-18f, -0.00363442646f, -0.00463495665f };
  for (int p = 0; p < 4; ++p)
    for (int lane = 0; lane < 32; ++lane) {
      int kb = (lane >> 4) << 1;
      int nn = lane & 15;
      for (int j = 0; j < 14; ++j) {
        int q = j >> 1, i = j & 1;
        int t = 4 * q + kb + i - nn;      // tap index c_p[t]
        int bi = 4 * t + 3 - p;           // index into BASE
        float val = 0.0f;
        if (t >= 0 && t < 12 && bi >= 0 && bi < 45) val = BASE[bi];
        a.v[p][lane][j] = val;
      }
      // [14],[15] stay zero
    }
  return a;
}

__constant__ AF AFRAG = make_af();

__global__ __launch_bounds__(256)
void upfir4x_wmma(const float* __restrict__ x, float* __restrict__ out) {
  __shared__ float xs[HALO * XSTR];   // padded input tile  (28 x 28)
  __shared__ float im[HALO * ISTR];   // W-filtered rows    (28 x 64)

  const int tid  = threadIdx.x;
  const int lane = tid & 31;
  const int wv   = tid >> 5;          // wave id, 0..7
  const int tx   = blockIdx.x & 7;    // tile col (64-wide output tiles)
  const int ty   = blockIdx.x >> 3;   // tile row
  const int img  = blockIdx.y;        // b*C + c

  const float* __restrict__ xin = x + (size_t)img * (128 * 128);
  float* __restrict__ o = out + (size_t)img * (512 * 512);

  // ---- load 28x28 input tile (zero halo) into LDS ----
  const int c0 = tx * 16 - 5;
  const int r0 = ty * 16 - 5;
  for (int idx = tid; idx < HALO * HALO; idx += 256) {
    int i = idx / HALO;
    int j = idx - i * HALO;
    int gr = r0 + i, gc = c0 + j;
    float v = 0.0f;
    if ((unsigned)gr < 128u && (unsigned)gc < 128u) v = xin[gr * 128 + gc];
    xs[i * XSTR + j] = v;
  }
  __syncthreads();

  const int half = lane >> 4;   // 0: lanes 0-15, 1: lanes 16-31
  const int kb   = half << 1;   // K base within a 4-chunk (A/B operand layout)
  const int nn   = lane & 15;   // A row (M) and B column (N) index

  // branch-free A-fragment fetch: 4 x b128 from the 8KB constant table
  auto load_a = [&](int p, float* a) {
    const float4* src = reinterpret_cast<const float4*>(&AFRAG.v[p][lane][0]);
    float4 t0 = src[0], t1 = src[1], t2 = src[2], t3 = src[3];
    a[0]  = t0.x; a[1]  = t0.y; a[2]  = t0.z; a[3]  = t0.w;
    a[4]  = t1.x; a[5]  = t1.y; a[6]  = t1.z; a[7]  = t1.w;
    a[8]  = t2.x; a[9]  = t2.y; a[10] = t2.z; a[11] = t2.w;
    a[12] = t3.x; a[13] = t3.y;
  };

  // ---- stage 1: W-direction polyphase filter (28 rows -> 28x64 in LDS) ----
  {
    const int p  = wv & 3;              // phase handled by this wave
    const int n0 = (wv & 4) ? 12 : 0;   // row group 0-15 / 12-27 (overlap benign)
    float a[14];
    load_a(p, a);
    v8f acc = {0.f, 0.f, 0.f, 0.f, 0.f, 0.f, 0.f, 0.f};
    const int row = n0 + nn;            // image row = B's N index
#pragma unroll
    for (int q = 0; q < 7; ++q) {
      v2f av; av.x = a[2 * q];  av.y = a[2 * q + 1];
      v2f bv; bv.x = xs[row * XSTR + 4 * q + kb];
              bv.y = xs[row * XSTR + 4 * q + kb + 1];
      acc = __builtin_amdgcn_wmma_f32_16x16x4_f32(false, av, false, bv,
                                                  (short)0, acc, false, false);
    }
    const int mh = half << 3;
#pragma unroll
    for (int v = 0; v < 8; ++v) {
      // D: M = v+mh (output W position), N = nn (row); col = 4*M + p
      im[row * ISTR + 4 * (v + mh) + p] = acc[v];
    }
  }
  __syncthreads();

  // ---- stage 2: H-direction filter; two phases per wave share B operands ----
  {
    const int pa = (wv & 1) << 1;       // phase 0 or 2
    const int pb = pa + 1;              // phase 1 or 3
    const int w0 = (wv >> 1) << 4;      // 16-column group: 0,16,32,48
    float aA[14], aB[14];
    load_a(pa, aA);
    load_a(pb, aB);
    v8f accA = {0.f, 0.f, 0.f, 0.f, 0.f, 0.f, 0.f, 0.f};
    v8f accB = {0.f, 0.f, 0.f, 0.f, 0.f, 0.f, 0.f, 0.f};
#pragma unroll
    for (int q = 0; q < 7; ++q) {
      int rr = 4 * q + kb;              // intermediate row = K index
      v2f bv; bv.x = im[rr * ISTR + w0 + nn];
              bv.y = im[(rr + 1) * ISTR + w0 + nn];
      v2f avA; avA.x = aA[2 * q]; avA.y = aA[2 * q + 1];
      v2f avB; avB.x = aB[2 * q]; avB.y = aB[2 * q + 1];
      // two independent accumulator chains -> WMMA latency hiding
      accA = __builtin_amdgcn_wmma_f32_16x16x4_f32(false, avA, false, bv,
                                                   (short)0, accA, false, false);
      accB = __builtin_amdgcn_wmma_f32_16x16x4_f32(false, avB, false, bv,
                                                   (short)0, accB, false, false);
    }
    const int mh  = half << 3;
    const int col = tx * 64 + w0 + nn;
#pragma unroll
    for (int v = 0; v < 8; ++v) {
      int rO = 4 * (ty * 16 + v + mh);  // output rows 4*(h0+M)+p
      o[(size_t)(rO + pa) * 512 + col] = accA[v];
      o[(size_t)(rO + pb) * 512 + col] = accB[v];
    }
  }
}

extern "C" void kernel_launch(void* const* d_in, const int* in_sizes, int n_in,
                              void* d_out, int out_size, void* d_ws, size_t ws_size,
                              hipStream_t stream) {
  (void)in_sizes; (void)n_in; (void)out_size; (void)d_ws; (void)ws_size;
  const float* x = (const float*)d_in[0];
  float* out = (float*)d_out;
  // 8x8 output tiles of 64x64 per image, 16*8 = 128 images
  dim3 grid(64, 128);
  dim3 block(256);
  upfir4x_wmma<<<grid, block, 0, stream>>>(x, out);
}